// SphericalHarmonicsFindPeaks_12567074308350
// MI455X (gfx1250) — compile-verified
//
#include <hip/hip_runtime.h>
#include <stdint.h>

// ---------------------------------------------------------------------------
// SphericalHarmonicsFindPeaks for MI455X (gfx1250, wave32)
//   f = signal . sh   (two 256 x 129024 GEMVs, memory bound: 264MB @ 23.3TB/s)
//   -> strict 2D local maxima -> exact sorted top-1024 -> cross suppression
// GEMV runs on V_WMMA_F32_16X16X4_F32 (full f32 precision, matrix pipe).
// sh streams are loaded with NT temporal hint so the 264MB stream does not
// evict the small L2-resident working set (f buffers + sort keys, ~3MB).
// Top-k via full bitonic sort of sortable u64 keys (L2 resident, ~1MB/proj).
// ---------------------------------------------------------------------------

#define NSH   256
#define GA    512
#define GB    252
#define NP    (GA * GB)          // 129024
#define TOPK  1024
#define SORTN 131072             // next pow2 >= NP
#define THR2F 6.0238910682299241e-04f   // (2*pi/256)^2

typedef float v2f __attribute__((ext_vector_type(2)));
typedef float v8f __attribute__((ext_vector_type(8)));
typedef unsigned long long u64;
typedef unsigned int u32;

// ---------------------------------------------------------------------------
// Kernel 1: f[proj][p] = sum_i signal[i] * sh[proj][i][p] via WMMA f32 16x16x4
// A-fragment: signal chunk of 4, replicated across all 16 rows (every lane of
// a half holds the two K values mapped to its VGPR slot, per the ISA layout:
// VGPR0 = K{0,2}, VGPR1 = K{1,3} split by lane half). B-fragment: 4x16 chunk
// of sh with the matching (vgpr, lane-half) -> K mapping. D rows are all
// identical; lanes 0..15 of VGPR0 carry the 16 outputs.
// Each wave owns 4 tiles (64 contiguous grid points); block = 8 waves = 512.
// ---------------------------------------------------------------------------
__global__ void __launch_bounds__(256)
k_gemv_wmma(const float* __restrict__ signal,
            const float* __restrict__ sh1,
            const float* __restrict__ sh2,
            float* __restrict__ fbuf)
{
    const float* __restrict__ sh = blockIdx.y ? sh2 : sh1;
    float* __restrict__ f = fbuf + (size_t)blockIdx.y * NP;

    __shared__ float lsig[NSH];
    if (threadIdx.x < NSH) lsig[threadIdx.x] = signal[threadIdx.x];
    __syncthreads();

    const int wave  = threadIdx.x >> 5;
    const int lane  = threadIdx.x & 31;
    const int half  = lane >> 4;          // lane half selects K offset (0 or 2)
    const int n     = lane & 15;          // column within 16-wide tile
    const int pbase = blockIdx.x * 512 + wave * 64;

    v8f acc0 = {}, acc1 = {}, acc2 = {}, acc3 = {};

    for (int k0 = 0; k0 < NSH; k0 += 4) {
        v2f a;
        a.x = lsig[k0 + 2 * half];        // VGPR0: K = 0 / 2
        a.y = lsig[k0 + 2 * half + 1];    // VGPR1: K = 1 / 3

        // single-use stream: non-temporal so it bypasses L2 retention
        const float* brow = sh + (size_t)(k0 + 2 * half) * NP + pbase + n;
        v2f b0, b1, b2, b3;
        b0.x = __builtin_nontemporal_load(brow + 0);
        b0.y = __builtin_nontemporal_load(brow + 0 + NP);
        b1.x = __builtin_nontemporal_load(brow + 16);
        b1.y = __builtin_nontemporal_load(brow + 16 + NP);
        b2.x = __builtin_nontemporal_load(brow + 32);
        b2.y = __builtin_nontemporal_load(brow + 32 + NP);
        b3.x = __builtin_nontemporal_load(brow + 48);
        b3.y = __builtin_nontemporal_load(brow + 48 + NP);

        acc0 = __builtin_amdgcn_wmma_f32_16x16x4_f32(false, a, false, b0, (short)0, acc0, false, false);
        acc1 = __builtin_amdgcn_wmma_f32_16x16x4_f32(false, a, false, b1, (short)0, acc1, false, false);
        acc2 = __builtin_amdgcn_wmma_f32_16x16x4_f32(false, a, false, b2, (short)0, acc2, false, false);
        acc3 = __builtin_amdgcn_wmma_f32_16x16x4_f32(false, a, false, b3, (short)0, acc3, false, false);
    }

    if (lane < 16) {                      // row M=0 lives in VGPR0, lanes 0-15
        f[pbase + n]      = acc0[0];
        f[pbase + 16 + n] = acc1[0];
        f[pbase + 32 + n] = acc2[0];
        f[pbase + 48 + n] = acc3[0];
    }
}

// ---------------------------------------------------------------------------
// Kernel 2: strict local-max mask + sortable key
// key = (enc(f) << 32) | (0xFFFFFFFF - p) for peaks, else (0 << 32) | inv(p).
// Descending u64 sort => f desc, ties by index asc (matches stable top_k);
// invalid entries (key32==0) follow all peaks in index-ascending order.
// ---------------------------------------------------------------------------
__device__ __forceinline__ u32 enc_f32(float v)
{
    u32 u = __float_as_uint(v);
    return (u & 0x80000000u) ? ~u : (u | 0x80000000u);
}

__global__ void __launch_bounds__(256)
k_keys(const float* __restrict__ fbuf, u64* __restrict__ keys)
{
    const int p    = blockIdx.x * 256 + threadIdx.x;
    const int proj = blockIdx.y;
    const float* __restrict__ f = fbuf + (size_t)proj * NP;
    u64* __restrict__ kk = keys + (size_t)proj * SORTN;

    u64 key = 0ull;                       // padding: sorts last
    if (p < NP) {
        const int a = p / GB;
        const int b = p - a * GB;
        const float v = f[p];
        bool row = (b >= 1) && (b <= GB - 2) && (v > f[p - 1])  && (v > f[p + 1]);
        bool col = (a >= 1) && (a <= GA - 2) && (v > f[p - GB]) && (v > f[p + GB]);
        u32 k32 = (row && col) ? enc_f32(v) : 0u;
        key = ((u64)k32 << 32) | (u64)(0xFFFFFFFFu - (u32)p);
    }
    kk[p] = key;
}

// ---------------------------------------------------------------------------
// Bitonic sort (descending) over 131072 u64 keys per projection.
// Swap rule: at (i, i|j), swap if (a < b) == ((i & k) == 0).
// ---------------------------------------------------------------------------
__global__ void __launch_bounds__(1024)
k_bitonic_first(u64* __restrict__ keys)          // k = 2 .. 2048 in LDS
{
    __shared__ u64 s[2048];
    u64* __restrict__ g = keys + (size_t)blockIdx.y * SORTN;
    const int base = blockIdx.x * 2048;
    const int t = threadIdx.x;

    s[t]        = g[base + t];
    s[t + 1024] = g[base + t + 1024];
    __syncthreads();

    for (int k = 2; k <= 2048; k <<= 1) {
        for (int j = k >> 1; j > 0; j >>= 1) {
            const int i  = (t & (j - 1)) | ((t & ~(j - 1)) << 1);
            const int gi = base + i;
            u64 a = s[i], b = s[i | j];
            const bool up = ((gi & k) == 0);
            if ((a < b) == up) { s[i] = b; s[i | j] = a; }
            __syncthreads();
        }
    }
    g[base + t]        = s[t];
    g[base + t + 1024] = s[t + 1024];
}

__global__ void __launch_bounds__(256)
k_bitonic_global(u64* __restrict__ keys, int k, int j)   // j >= 2048
{
    u64* __restrict__ g = keys + (size_t)blockIdx.y * SORTN;
    const int t = blockIdx.x * 256 + threadIdx.x;        // [0, SORTN/2)
    const int i = (t & (j - 1)) | ((t & ~(j - 1)) << 1);
    u64 a = g[i], b = g[i | j];
    const bool up = ((i & k) == 0);
    if ((a < b) == up) { g[i] = b; g[i | j] = a; }
}

__global__ void __launch_bounds__(1024)
k_bitonic_last(u64* __restrict__ keys, int k)    // j = 1024 .. 1 in LDS
{
    __shared__ u64 s[2048];
    u64* __restrict__ g = keys + (size_t)blockIdx.y * SORTN;
    const int base = blockIdx.x * 2048;
    const int t = threadIdx.x;

    s[t]        = g[base + t];
    s[t + 1024] = g[base + t + 1024];
    __syncthreads();

    for (int j = 1024; j > 0; j >>= 1) {
        const int i  = (t & (j - 1)) | ((t & ~(j - 1)) << 1);
        const int gi = base + i;
        u64 a = s[i], b = s[i | j];
        const bool up = ((gi & k) == 0);
        if ((a < b) == up) { s[i] = b; s[i | j] = a; }
        __syncthreads();
    }
    g[base + t]        = s[t];
    g[base + t + 1024] = s[t + 1024];
}

// ---------------------------------------------------------------------------
// Kernel 4: materialize top-1024 (peaks, radius, valid) per projection.
// ---------------------------------------------------------------------------
__global__ void __launch_bounds__(256)
k_extract(const u64* __restrict__ keys,
          const float* __restrict__ fbuf,
          const float* __restrict__ xyz1,
          const float* __restrict__ xyz2,
          float* __restrict__ pk,     // [2][TOPK][3]
          float* __restrict__ rad,    // [2][TOPK]
          float* __restrict__ val)    // [2][TOPK]
{
    const int t    = blockIdx.x * 256 + threadIdx.x;     // [0, TOPK)
    const int proj = blockIdx.y;
    const u64 e = keys[(size_t)proj * SORTN + t];
    const u32 k32 = (u32)(e >> 32);
    const u32 idx = 0xFFFFFFFFu - (u32)(e & 0xFFFFFFFFu);
    const bool v = (k32 != 0u);
    const float* __restrict__ xyz = proj ? xyz2 : xyz1;

    float r = 0.f, x = 0.f, y = 0.f, z = 0.f;
    if (v) {
        r = fbuf[(size_t)proj * NP + idx];
        x = xyz[3 * idx];
        y = xyz[3 * idx + 1];
        z = xyz[3 * idx + 2];
    }
    const size_t o = (size_t)proj * TOPK + t;
    pk[3 * o]     = x;
    pk[3 * o + 1] = y;
    pk[3 * o + 2] = z;
    rad[o] = r;
    val[o] = v ? 1.f : 0.f;
}

// ---------------------------------------------------------------------------
// Kernel 5: suppress proj-1 peaks within THR of any valid proj-2 peak;
// write concatenated output: peaks[2048][3] | radius[2048] | valid[2048].
// ---------------------------------------------------------------------------
__global__ void __launch_bounds__(256)
k_final(const float* __restrict__ pk,
        const float* __restrict__ rad,
        const float* __restrict__ val,
        float* __restrict__ out)
{
    const int t = blockIdx.x * 256 + threadIdx.x;        // [0, 2*TOPK)
    float* __restrict__ opk  = out;                      // 6144 floats
    float* __restrict__ orad = out + 2 * TOPK * 3;       // 2048 floats
    float* __restrict__ oval = orad + 2 * TOPK;          // 2048 floats
    const float* __restrict__ p2 = pk + TOPK * 3;

    if (t < TOPK) {
        const bool v1 = (val[t] != 0.f);
        const float x = pk[3 * t], y = pk[3 * t + 1], z = pk[3 * t + 2];
        bool keep = v1;
        if (v1) {
            for (int u = 0; u < TOPK; ++u) {
                if (val[TOPK + u] != 0.f) {
                    const float dx = x - p2[3 * u];
                    const float dy = y - p2[3 * u + 1];
                    const float dz = z - p2[3 * u + 2];
                    if (dx * dx + dy * dy + dz * dz < THR2F) { keep = false; break; }
                }
            }
        }
        opk[3 * t]     = keep ? x : 0.f;
        opk[3 * t + 1] = keep ? y : 0.f;
        opk[3 * t + 2] = keep ? z : 0.f;
        orad[t] = keep ? rad[t] : 0.f;
        oval[t] = keep ? 1.f : 0.f;
    } else {
        const int u = t - TOPK;
        opk[3 * t]     = p2[3 * u];
        opk[3 * t + 1] = p2[3 * u + 1];
        opk[3 * t + 2] = p2[3 * u + 2];
        orad[t] = rad[TOPK + u];
        oval[t] = val[TOPK + u];
    }
}

// ---------------------------------------------------------------------------
extern "C" void kernel_launch(void* const* d_in, const int* in_sizes, int n_in,
                              void* d_out, int out_size, void* d_ws, size_t ws_size,
                              hipStream_t stream)
{
    const float* signal = (const float*)d_in[0];   // [256]
    const float* sh1    = (const float*)d_in[1];   // [256][512][252]
    const float* sh2    = (const float*)d_in[2];   // [256][512][252]
    const float* xyz1   = (const float*)d_in[3];   // [512][252][3]
    const float* xyz2   = (const float*)d_in[4];   // [512][252][3]
    float* out = (float*)d_out;                    // 10240 floats

    // workspace carve-up (~3.1 MB)
    char* ws = (char*)d_ws;
    float* fbuf = (float*)ws;                                   // 2*NP f32
    u64*   keys = (u64*)(ws + (size_t)2 * NP * sizeof(float));  // 2*SORTN u64 (8B aligned)
    float* pk   = (float*)((char*)keys + (size_t)2 * SORTN * sizeof(u64));
    float* rad  = pk + 2 * TOPK * 3;
    float* val  = rad + 2 * TOPK;

    // 1) dual GEMV on the WMMA pipe (memory-bound: streams both sh tensors once)
    k_gemv_wmma<<<dim3(NP / 512, 2), 256, 0, stream>>>(signal, sh1, sh2, fbuf);

    // 2) peak mask + sortable keys
    k_keys<<<dim3(SORTN / 256, 2), 256, 0, stream>>>(fbuf, keys);

    // 3) bitonic sort, descending (L2-resident, 28 passes)
    k_bitonic_first<<<dim3(SORTN / 2048, 2), 1024, 0, stream>>>(keys);
    for (int k = 4096; k <= SORTN; k <<= 1) {
        for (int j = k >> 1; j >= 2048; j >>= 1)
            k_bitonic_global<<<dim3(SORTN / 512, 2), 256, 0, stream>>>(keys, k, j);
        k_bitonic_last<<<dim3(SORTN / 2048, 2), 1024, 0, stream>>>(keys, k);
    }

    // 4) top-1024 per projection
    k_extract<<<dim3(TOPK / 256, 2), 256, 0, stream>>>(keys, fbuf, xyz1, xyz2, pk, rad, val);

    // 5) cross suppression + packed output
    k_final<<<dim3(2 * TOPK / 256), 256, 0, stream>>>(pk, rad, val, out);
}